// InferModule_18227841204320
// MI455X (gfx1250) — compile-verified
//
#include <hip/hip_runtime.h>
#include <math.h>

// Problem constants (match reference setup_inputs)
#define B_ 32
#define G_ 2048
#define C_ 16
#define S_ 8
#define L_ 3
#define M_ 16
#define STEPS 5
#define GAMMA 0.01f
#define INV_GAMMA 100.0f

typedef __attribute__((ext_vector_type(2))) float v2f;
typedef __attribute__((ext_vector_type(8))) float v8f;

// All softor intermediates here are >= 0, so float max == uint max on the raw bits.
__device__ __forceinline__ void atomicMaxNonneg(unsigned* p, float v) {
    atomicMax(p, __float_as_uint(v));
}

// ---------------------------------------------------------------------------
// init: Rbuf = x ; Ws = softmax(W, axis=1) ; mx[0..2] = 0
// ---------------------------------------------------------------------------
__global__ __launch_bounds__(256) void initKernel(const float* __restrict__ x,
                                                  const float* __restrict__ W,
                                                  float* __restrict__ Rbuf,
                                                  float* __restrict__ Ws,
                                                  unsigned* __restrict__ mx) {
    int t = blockIdx.x * blockDim.x + threadIdx.x;
    if (t < B_ * G_) Rbuf[t] = x[t];
    if (blockIdx.x == 0) {
        int tid = threadIdx.x;
        if (tid < M_) {
            const float* wr = W + tid * C_;
            float mw = wr[0];
            #pragma unroll
            for (int c = 1; c < C_; ++c) mw = fmaxf(mw, wr[c]);
            float sum = 0.f;
            #pragma unroll
            for (int c = 0; c < C_; ++c) {
                float e = __expf(wr[c] - mw);
                Ws[tid * C_ + c] = e;
                sum += e;
            }
            float inv = 1.0f / sum;
            #pragma unroll
            for (int c = 0; c < C_; ++c) Ws[tid * C_ + c] *= inv;
        }
        if (tid == M_) { mx[0] = 0u; mx[1] = 0u; mx[2] = 0u; }
    }
}

// ---------------------------------------------------------------------------
// gather: y1[b,c,g] = gamma*logsumexp_s( prod_l R[b, I[c,g,s,l]] / gamma )
// atomics into mx[0]; resets mx[2] (consumed by previous normalize)
// ---------------------------------------------------------------------------
__global__ __launch_bounds__(256) void gatherKernel(const float* __restrict__ Rbuf,
                                                    const int* __restrict__ I,
                                                    float* __restrict__ y1,
                                                    unsigned* __restrict__ mx) {
    int t = blockIdx.x * blockDim.x + threadIdx.x;   // (b*C + c)*G + g, g fastest
    if (t >= B_ * C_ * G_) return;
    int g  = t & (G_ - 1);
    int bc = t >> 11;           // G_ == 2048
    int c  = bc & (C_ - 1);
    int b  = bc >> 4;           // C_ == 16

    const int*   idx  = I + (c * G_ + g) * (S_ * L_);  // 24 contiguous ints
    const float* Rrow = Rbuf + b * G_;

    float pv[S_];
    float mp = -INFINITY;
    #pragma unroll
    for (int s = 0; s < S_; ++s) {
        float p = 1.0f;
        #pragma unroll
        for (int l = 0; l < L_; ++l) p *= Rrow[idx[s * L_ + l]];
        pv[s] = p;
        mp = fmaxf(mp, p);
    }
    float sum = 0.f;
    #pragma unroll
    for (int s = 0; s < S_; ++s) sum += __expf((pv[s] - mp) * INV_GAMMA);
    float y = mp + GAMMA * __logf(sum);

    y1[t] = y;
    atomicMaxNonneg(&mx[0], y);
    if (t == 0) mx[2] = 0u;
}

// ---------------------------------------------------------------------------
// matmul (WMMA): Cv = y1 * scale1 ; H = Ws(16x16) x Cv(b,:,gtile) (16x16 tile)
// via 4 chained V_WMMA_F32_16X16X4_F32 ; then y2[b,g] = gamma*logsumexp_m(H/gamma)
// One wave per (b, gtile). atomics into mx[1].
// A layout (16x4 f32):  VGPR j, lane l: M = l&15, K = j + 2*(l>=16)   [ISA table]
// B layout (4x16 f32):  VGPR j, lane l: K = j + 2*(l>=16), N = l&15   [mirror]
// D layout (16x16 f32): VGPR r, lane l: M = r + 8*(l>=16), N = l&15   [ISA table]
// ---------------------------------------------------------------------------
__global__ __launch_bounds__(256) void matmulKernel(const float* __restrict__ y1,
                                                    const float* __restrict__ Ws,
                                                    float* __restrict__ y2,
                                                    unsigned* __restrict__ mx) {
    const int lane = threadIdx.x & 31;
    const int wave = threadIdx.x >> 5;
    const int tile = blockIdx.x * 8 + wave;          // 4096 tiles total
    const int b    = tile >> 7;                      // / (G_/16)
    const int g0   = (tile & 127) << 4;

    const float mx1 = __uint_as_float(mx[0]);
    const float s1  = (mx1 > 1.0f) ? (1.0f / mx1) : 1.0f;

    const int half = lane >> 4;        // 0 or 1
    const int ln   = lane & 15;

    v8f acc = {};
    #pragma unroll
    for (int k = 0; k < 4; ++k) {
        const int k0 = 4 * k + 2 * half;
        v2f a, bb;
        a.x = Ws[ln * C_ + (k0 + 0)];
        a.y = Ws[ln * C_ + (k0 + 1)];
        bb.x = y1[(b * C_ + (k0 + 0)) * G_ + g0 + ln] * s1;
        bb.y = y1[(b * C_ + (k0 + 1)) * G_ + g0 + ln] * s1;
        acc = __builtin_amdgcn_wmma_f32_16x16x4_f32(
            /*neg_a=*/false, a, /*neg_b=*/false, bb,
            /*c_mod=*/(short)0, acc, /*reuse_a=*/false, /*reuse_b=*/false);
    }

    // logsumexp over m: this lane holds m = r + 8*half (r = 0..7);
    // partner lane (lane ^ 16) holds the other 8 m values.
    float hmax = acc[0];
    #pragma unroll
    for (int r = 1; r < 8; ++r) hmax = fmaxf(hmax, acc[r]);
    hmax = fmaxf(hmax, __shfl_xor(hmax, 16, 32));

    float sum = 0.f;
    #pragma unroll
    for (int r = 0; r < 8; ++r) sum += __expf((acc[r] - hmax) * INV_GAMMA);
    sum += __shfl_xor(sum, 16, 32);

    float y = hmax + GAMMA * __logf(sum);
    if (half == 0) {
        y2[b * G_ + g0 + ln] = y;
        atomicMaxNonneg(&mx[1], y);
    }
}

// ---------------------------------------------------------------------------
// combine: Hn = y2*scale2 ; y2 <- gamma*logsumexp([R, Hn]/gamma) elementwise
// atomics into mx[2]; resets mx[0]
// ---------------------------------------------------------------------------
__global__ __launch_bounds__(256) void combineKernel(const float* __restrict__ Rbuf,
                                                     float* __restrict__ y2,
                                                     unsigned* __restrict__ mx) {
    int t = blockIdx.x * blockDim.x + threadIdx.x;
    if (t >= B_ * G_) return;
    const float mx2 = __uint_as_float(mx[1]);
    const float s2  = (mx2 > 1.0f) ? (1.0f / mx2) : 1.0f;
    float h = y2[t] * s2;
    float a = Rbuf[t];
    float m = fmaxf(a, h);
    float y = m + GAMMA * __logf(__expf((a - m) * INV_GAMMA) +
                                 __expf((h - m) * INV_GAMMA));
    y2[t] = y;
    atomicMaxNonneg(&mx[2], y);
    if (t == 0) mx[0] = 0u;
}

// ---------------------------------------------------------------------------
// normalize: R = (mx3 > 1) ? y2/mx3 : y2 ; optionally emit final output.
// resets mx[1]
// ---------------------------------------------------------------------------
__global__ __launch_bounds__(256) void normalizeKernel(float* __restrict__ Rbuf,
                                                       const float* __restrict__ y2,
                                                       unsigned* __restrict__ mx,
                                                       float* __restrict__ out) {
    int t = blockIdx.x * blockDim.x + threadIdx.x;
    if (t >= B_ * G_) return;
    const float mx3 = __uint_as_float(mx[2]);
    const float s3  = (mx3 > 1.0f) ? (1.0f / mx3) : 1.0f;
    float r = y2[t] * s3;
    Rbuf[t] = r;
    if (out) out[t] = r;
    if (t == 0) mx[1] = 0u;
}

// ---------------------------------------------------------------------------
extern "C" void kernel_launch(void* const* d_in, const int* in_sizes, int n_in,
                              void* d_out, int out_size, void* d_ws, size_t ws_size,
                              hipStream_t stream) {
    const float* x = (const float*)d_in[0];   // (B, G)
    const float* W = (const float*)d_in[1];   // (m, C)
    const int*   I = (const int*)d_in[2];     // (C, G, S, L)
    float* out = (float*)d_out;               // (B, G)

    // workspace carve-up
    float* y1  = (float*)d_ws;                 // B*C*G
    float* y2  = y1 + (size_t)B_ * C_ * G_;    // B*G
    float* Rb  = y2 + (size_t)B_ * G_;         // B*G
    float* Ws  = Rb + (size_t)B_ * G_;         // m*C
    unsigned* mx = (unsigned*)(Ws + M_ * C_);  // 3 scalars

    initKernel<<<(B_ * G_) / 256, 256, 0, stream>>>(x, W, Rb, Ws, mx);

    for (int step = 0; step < STEPS; ++step) {
        gatherKernel<<<(B_ * C_ * G_) / 256, 256, 0, stream>>>(Rb, I, y1, mx);
        matmulKernel<<<(B_ * (G_ / 16)) / 8, 256, 0, stream>>>(y1, Ws, y2, mx);
        combineKernel<<<(B_ * G_) / 256, 256, 0, stream>>>(Rb, y2, mx);
        normalizeKernel<<<(B_ * G_) / 256, 256, 0, stream>>>(
            Rb, y2, mx, (step == STEPS - 1) ? out : nullptr);
    }
}